// SingleConv_63376537420065
// MI455X (gfx1250) — compile-verified
//
#include <hip/hip_runtime.h>
#include <hip/hip_bf16.h>

typedef __attribute__((ext_vector_type(2))) float v2f;
typedef __attribute__((ext_vector_type(8))) float v8f;

#define HSTR 36   // h_lds row stride (dwords): conflict-free A-frag b64 reads
#define BSTR 40   // basis_lds row stride (dwords): 160B rows -> 16B-aligned b96 reads

// ---------------------------------------------------------------------------
// Kernel 1: radial MLP  (Linear->LN->ReLU->Linear->LN->ReLU), h -> workspace
// One edge per lane; weights staged in LDS, read as lane-broadcasts.
// ---------------------------------------------------------------------------
__global__ __launch_bounds__(128) void mlp_kernel(
    const float* __restrict__ f,
    const float* __restrict__ w1, const float* __restrict__ b1,
    const float* __restrict__ g1, const float* __restrict__ be1,
    const float* __restrict__ w2, const float* __restrict__ b2,
    const float* __restrict__ g2, const float* __restrict__ be2,
    float* __restrict__ hws, int E)
{
  __shared__ float wl[1760];   // w1[0,544) b1 g1 be1 | w2[640,1664) b2 g2 be2
  const int tid = threadIdx.x;
  for (int i = tid; i < 544; i += 128) wl[i] = w1[i];
  if (tid < 32) {
    wl[544 + tid] = b1[tid];  wl[576 + tid] = g1[tid];  wl[608 + tid] = be1[tid];
    wl[1664 + tid] = b2[tid]; wl[1696 + tid] = g2[tid]; wl[1728 + tid] = be2[tid];
  }
  for (int i = tid; i < 1024; i += 128) wl[640 + i] = w2[i];
  __syncthreads();

  const int e = blockIdx.x * 128 + tid;
  if (e >= E) return;

  float fr[17];
#pragma unroll
  for (int k = 0; k < 17; ++k) fr[k] = f[(size_t)e * 17 + k];

  float x[32];
#pragma unroll
  for (int c = 0; c < 32; ++c) {
    float a = wl[544 + c];
#pragma unroll
    for (int k = 0; k < 17; ++k) a = fmaf(fr[k], wl[c * 17 + k], a);
    x[c] = a;
  }
  // LN + ReLU (per-lane: all 32 channels live in this lane's VGPRs)
  {
    float mu = 0.f;
#pragma unroll
    for (int c = 0; c < 32; ++c) mu += x[c];
    mu *= 0.03125f;
    float var = 0.f;
#pragma unroll
    for (int c = 0; c < 32; ++c) { float d = x[c] - mu; var = fmaf(d, d, var); }
    var *= 0.03125f;
    float rs = rsqrtf(var + 1e-5f);
#pragma unroll
    for (int c = 0; c < 32; ++c) {
      float xn = (x[c] - mu) * rs * wl[576 + c] + wl[608 + c];
      x[c] = fmaxf(xn, 0.f);
    }
  }
  // Layer 2
  float y[32];
#pragma unroll
  for (int c = 0; c < 32; ++c) {
    float a = wl[1664 + c];
#pragma unroll
    for (int k = 0; k < 32; ++k) a = fmaf(x[k], wl[640 + c * 32 + k], a);
    y[c] = a;
  }
  {
    float mu = 0.f;
#pragma unroll
    for (int c = 0; c < 32; ++c) mu += y[c];
    mu *= 0.03125f;
    float var = 0.f;
#pragma unroll
    for (int c = 0; c < 32; ++c) { float d = y[c] - mu; var = fmaf(d, d, var); }
    var *= 0.03125f;
    float rs = rsqrtf(var + 1e-5f);
#pragma unroll
    for (int c = 0; c < 32; ++c) {
      float yn = (y[c] - mu) * rs * wl[1696 + c] + wl[1728 + c];
      y[c] = fmaxf(yn, 0.f);
    }
  }
  float4* hp = (float4*)(hws + (size_t)e * 32);
#pragma unroll
  for (int q = 0; q < 8; ++q)
    hp[q] = make_float4(y[q * 4], y[q * 4 + 1], y[q * 4 + 2], y[q * 4 + 3]);
}

// ---------------------------------------------------------------------------
// Kernel 2: r = h @ w3.T + b3 via V_WMMA_F32_16X16X4_F32, fused with
// kernel[e, co*3+do, ci*3+di] = sum_f R[e,co,ci,f] * B[e,do,di,f]
// Block = 8 waves, 32 edges (2 WMMA tiles of 16). Wave w owns column groups
// [8w, 8w+8): group g -> co = g>>1, ci-half = (g&1)*16, 3 freq accumulators.
// ---------------------------------------------------------------------------
__global__ __launch_bounds__(256) void conv_kernel(
    const float* __restrict__ hws, const float* __restrict__ basis,
    const float* __restrict__ w3, const float* __restrict__ b3,
    float* __restrict__ out, int E)
{
  __shared__ float h_lds[32 * HSTR];
  __shared__ float bas_lds[32 * BSTR];
  const int tid  = threadIdx.x;
  const int eblk = blockIdx.x * 32;

  // Stage h tile (32 edges x 32 ch) into LDS, stride 36
  {
    const int idx = tid * 4;           // 1024 floats / 256 threads
    const int le = idx >> 5, k = idx & 31;
    const int ge = eblk + le;
    float4 hv = (ge < E) ? *(const float4*)(hws + (size_t)ge * 32 + k)
                         : make_float4(0.f, 0.f, 0.f, 0.f);
    float* dst = &h_lds[le * HSTR + k];
    dst[0] = hv.x; dst[1] = hv.y; dst[2] = hv.z; dst[3] = hv.w;
  }
  // Stage basis (32 edges x 27) into LDS as [le][do*3+di][f], stride 4 per (do,di)
  for (int idx = tid; idx < 32 * 27; idx += 256) {
    const int le = idx / 27, j = idx - le * 27;
    const int ge = eblk + le;
    float v = (ge < E) ? basis[(size_t)ge * 27 + j] : 0.f;
    bas_lds[le * BSTR + (j / 3) * 4 + (j % 3)] = v;
  }
  __syncthreads();

  const int lane = tid & 31;           // wave32
  const int wave = tid >> 5;
  const int n  = lane & 15;            // B-matrix column / A-matrix row index
  const int hi = lane >> 4;            // half-wave: K offset 2*hi

  for (int gi = 0; gi < 8; ++gi) {
    const int g   = wave * 8 + gi;
    const int co  = g >> 1;
    const int cih = (g & 1) << 4;
    const int colrow = (co * 32 + cih + n) * 3;   // w3 row for freq f = colrow + f

    // B fragments: bf[f][s] covers K = [4s, 4s+4), N = 16 cols of this group
    v2f bf[3][8];
    const float* w3p = w3 + (size_t)colrow * 32 + 2 * hi;
#pragma unroll
    for (int fq = 0; fq < 3; ++fq)
#pragma unroll
      for (int s = 0; s < 8; ++s)
        bf[fq][s] = *(const v2f*)(w3p + fq * 32 + 4 * s);

    const float b3v0 = b3[colrow + 0];
    const float b3v1 = b3[colrow + 1];
    const float b3v2 = b3[colrow + 2];

#pragma unroll
    for (int t = 0; t < 2; ++t) {
      v8f c0 = {0.f,0.f,0.f,0.f,0.f,0.f,0.f,0.f};
      v8f c1 = {0.f,0.f,0.f,0.f,0.f,0.f,0.f,0.f};
      v8f c2 = {0.f,0.f,0.f,0.f,0.f,0.f,0.f,0.f};
      const float* hp = &h_lds[(t * 16 + n) * HSTR + 2 * hi];
#pragma unroll
      for (int s = 0; s < 8; ++s) {
        v2f a = *(const v2f*)(hp + 4 * s);   // conflict-free ds_load_b64
        c0 = __builtin_amdgcn_wmma_f32_16x16x4_f32(false, a, false, bf[0][s],
                                                   (short)0, c0, false, false);
        c1 = __builtin_amdgcn_wmma_f32_16x16x4_f32(false, a, false, bf[1][s],
                                                   (short)0, c1, false, false);
        c2 = __builtin_amdgcn_wmma_f32_16x16x4_f32(false, a, false, bf[2][s],
                                                   (short)0, c2, false, false);
      }
      // Epilogue: per C-VGPR v, edge = t*16 + v + 8*hi (shared by 16 lanes of
      // the half -> bas_lds reads are broadcasts). 16 lanes write 48
      // contiguous floats per (do) row; streaming NT stores.
#pragma unroll
      for (int v = 0; v < 8; ++v) {
        const int eloc = t * 16 + v + hi * 8;
        const int eg   = eblk + eloc;
        const float r0 = c0[v] + b3v0;
        const float r1 = c1[v] + b3v1;
        const float r2 = c2[v] + b3v2;
        const float* bp = &bas_lds[eloc * BSTR];
        if (eg < E) {
          float* op = out + (size_t)eg * 9216 + (co * 3) * 96 + (cih + n) * 3;
#pragma unroll
          for (int dd = 0; dd < 3; ++dd)
#pragma unroll
            for (int di = 0; di < 3; ++di) {
              const float* bq = bp + (dd * 3 + di) * 4;
              float o = fmaf(r2, bq[2], fmaf(r1, bq[1], r0 * bq[0]));
              __builtin_nontemporal_store(o, op + dd * 96 + di);
            }
        }
      }
    }
  }
}

// ---------------------------------------------------------------------------
extern "C" void kernel_launch(void* const* d_in, const int* in_sizes, int n_in,
                              void* d_out, int out_size, void* d_ws, size_t ws_size,
                              hipStream_t stream) {
  const float* f    = (const float*)d_in[0];
  const float* bas  = (const float*)d_in[1];
  const float* w1   = (const float*)d_in[2];
  const float* b1   = (const float*)d_in[3];
  const float* g1   = (const float*)d_in[4];
  const float* be1  = (const float*)d_in[5];
  const float* w2   = (const float*)d_in[6];
  const float* b2   = (const float*)d_in[7];
  const float* g2   = (const float*)d_in[8];
  const float* be2  = (const float*)d_in[9];
  const float* w3   = (const float*)d_in[10];
  const float* b3   = (const float*)d_in[11];
  float* out = (float*)d_out;
  float* hws = (float*)d_ws;               // E*32 floats = 2.56 MB scratch

  const int E = in_sizes[0] / 17;
  mlp_kernel<<<(E + 127) / 128, 128, 0, stream>>>(f, w1, b1, g1, be1,
                                                  w2, b2, g2, be2, hws, E);
  conv_kernel<<<(E + 31) / 32, 256, 0, stream>>>(hws, bas, w3, b3, out, E);
}